// LearnedOptimizer_55164559950083
// MI455X (gfx1250) — compile-verified
//
#include <hip/hip_runtime.h>
#include <hip/hip_bf16.h>
#include <math.h>

// Learned-optimizer single LSTM step, fused one-pass streaming kernel.
// Memory-bound (~1.08 GB traffic -> ~46us @ 23.3 TB/s HBM). Gates matmul
// ([N,8]@[8,24]) runs on V_WMMA_F32_16X16X4_F32 (fp32, exact); all bulk
// traffic uses non-temporal hints since every element is touched once.

typedef float v2f __attribute__((ext_vector_type(2)));
typedef float v8f __attribute__((ext_vector_type(8)));

#define LO_INV_P   0.1f                      // 1/P, P=10
#define LO_THRESH  4.539992976248485e-05f    // exp(-10) (> f32 tiny)
#define LO_EXP_P   22026.465794806718f       // exp(10)
#define LO_TINY    1.1754943508222875e-38f   // f32 tiny
#define GROUPS     8                         // 32-param groups per wave

__device__ __forceinline__ float lo_sigmoid(float x) {
    return 1.0f / (1.0f + __expf(-x));
}

// W_cat[row][k]: k<2 -> W_ih[row][k] ([24,2] row-major), k>=2 -> W_hh[row][k-2] ([24,6])
__device__ __forceinline__ float lo_getW(const float* __restrict__ W_ih,
                                         const float* __restrict__ W_hh,
                                         int row, int k) {
    if (row >= 24) return 0.0f;
    return (k < 2) ? W_ih[row * 2 + k] : W_hh[row * 6 + (k - 2)];
}

// Branchless rule-of-thumb gradient preprocessing.
__device__ __forceinline__ void lo_preprocess(float g, float& mag, float& sgn) {
    float ag  = fabsf(g);
    float lg  = __logf(ag + LO_TINY) * LO_INV_P;      // always safe: arg > 0
    float sg  = (g > 0.0f) ? 1.0f : ((g < 0.0f) ? -1.0f : 0.0f);
    bool  big = ag >= LO_THRESH;
    mag = big ? lg : -1.0f;
    sgn = big ? sg : g * LO_EXP_P;
}

__device__ __forceinline__ float lo_ldnt(const float* p) {
    return __builtin_nontemporal_load(p);
}
__device__ __forceinline__ v2f lo_ldnt2(const float* p) {
    return __builtin_nontemporal_load((const v2f*)p);
}
__device__ __forceinline__ void lo_stnt2(float* p, float a, float b) {
    v2f v; v.x = a; v.y = b;
    __builtin_nontemporal_store(v, (v2f*)p);
}

__global__ void lopt_lstm_step_kernel(
    const float* __restrict__ p1, const float* __restrict__ p2,
    const float* __restrict__ g1, const float* __restrict__ g2,
    const float* __restrict__ h0, const float* __restrict__ c0,
    const float* __restrict__ W_ih, const float* __restrict__ W_hh,
    const float* __restrict__ b_ih, const float* __restrict__ b_hh,
    const float* __restrict__ W_out, const float* __restrict__ b_out,
    float* __restrict__ out, long long n1, long long N)
{
    const int lane = threadIdx.x & 31;
    const int half = lane >> 4;          // 0: lanes 0-15, 1: lanes 16-31
    const int pl   = lane & 15;
    const long long wid = ((long long)blockIdx.x * blockDim.x + threadIdx.x) >> 5;

    // ---- per-wave constant WMMA operands ---------------------------------
    // A fragments: weight slab, A[gt][kc], rows = gates gt*16+pl, cols K.
    // 16x4 f32 A layout: lanes 0-15 hold K = kc*4 + {0,1}; lanes 16-31 K = kc*4 + {2,3}.
    v2f A[2][2];
#pragma unroll
    for (int gt = 0; gt < 2; ++gt) {
#pragma unroll
        for (int kc = 0; kc < 2; ++kc) {
            int row = pl + gt * 16;
            int kb  = kc * 4 + half * 2;
            A[gt][kc].x = lo_getW(W_ih, W_hh, row, kb + 0);
            A[gt][kc].y = lo_getW(W_ih, W_hh, row, kb + 1);
        }
    }
    // C bias fragments: D rows = gates; VGPR v at this lane is gate gt*16 + half*8 + v.
    v8f CB[2];
#pragma unroll
    for (int gt = 0; gt < 2; ++gt) {
#pragma unroll
        for (int v = 0; v < 8; ++v) {
            int j = gt * 16 + half * 8 + v;
            CB[gt][v] = (j < 24) ? (b_ih[j] + b_hh[j]) : 0.0f;
        }
    }
    float wo[6];
#pragma unroll
    for (int k = 0; k < 6; ++k) wo[k] = W_out[k];
    const float bo = b_out[0];

    for (int it = 0; it < GROUPS; ++it) {
        const long long pbase = (wid * GROUPS + it) * 32;   // uniform per wave
        if (pbase >= N) break;

        float gate[24];
        long long myp;
        bool full = (pbase + 32 <= N);                      // uniform per wave

        if (full) {
            // ---- full group: 32 params = two 16-param B tiles -------------
            v8f D[2][2]; // D[param_tile][gate_tile]
#pragma unroll
            for (int pt = 0; pt < 2; ++pt) {
                const long long pp = pbase + pt * 16 + pl;  // param column for this lane
                // B fragments: features [mag,sign,h0..h5] of param pp,
                // same K striping as A: half0 -> K {kc*4+0,+1}, half1 -> {+2,+3}.
                v2f B0, B1;
                if (half == 0) {
                    float gg = lo_ldnt((pp < n1) ? (g1 + pp) : (g2 + (pp - n1)));
                    float mag, sgn;
                    lo_preprocess(gg, mag, sgn);             // K=0,1
                    B0.x = mag; B0.y = sgn;
                    B1 = lo_ldnt2(h0 + pp * 6 + 2);          // K=4,5 -> h2,h3
                } else {
                    B0 = lo_ldnt2(h0 + pp * 6 + 0);          // K=2,3 -> h0,h1
                    B1 = lo_ldnt2(h0 + pp * 6 + 4);          // K=6,7 -> h4,h5
                }
#pragma unroll
                for (int gt = 0; gt < 2; ++gt) {
                    v8f acc = __builtin_amdgcn_wmma_f32_16x16x4_f32(
                        false, A[gt][0], false, B0, (short)0, CB[gt], false, false);
                    acc = __builtin_amdgcn_wmma_f32_16x16x4_f32(
                        false, A[gt][1], false, B1, (short)0, acc, false, false);
                    D[pt][gt] = acc;
                }
            }
            // Cross-half swap: lane {half,pl} gathers all 24 gates of param
            // pbase + half*16 + pl.  Two xor-16 swaps per VGPR suffice:
            //  t: lower lanes send tile1 gates0..7, upper lanes send tile0 gates8..15
            //  u: lower lanes send tile1 gates16..23 (only upper half consumes)
#pragma unroll
            for (int v = 0; v < 8; ++v) {
                float d00 = D[0][0][v];
                float d01 = D[0][1][v];
                float d10 = D[1][0][v];
                float d11 = D[1][1][v];
                float t  = half ? d00 : d10;
                float tx = __shfl_xor(t, 16, 32);
                float ux = __shfl_xor(d11, 16, 32);
                gate[v]      = half ? tx  : d00;   // gates 0..7
                gate[8 + v]  = half ? d10 : tx;    // gates 8..15
                gate[16 + v] = half ? ux  : d01;   // gates 16..23
            }
            myp = pbase + (long long)half * 16 + pl;
        } else {
            // ---- scalar tail fallback (uniform branch; no WMMA) -----------
            myp = pbase + lane;
            if (myp >= N) continue;
            float feat[8];
            float gg = (myp < n1) ? g1[myp] : g2[myp - n1];
            lo_preprocess(gg, feat[0], feat[1]);
#pragma unroll
            for (int k = 0; k < 6; ++k) feat[2 + k] = h0[myp * 6 + k];
#pragma unroll
            for (int j = 0; j < 24; ++j) {
                float s = b_ih[j] + b_hh[j];
                s += feat[0] * W_ih[j * 2 + 0];
                s += feat[1] * W_ih[j * 2 + 1];
#pragma unroll
                for (int k = 0; k < 6; ++k) s += feat[2 + k] * W_hh[j * 6 + k];
                gate[j] = s;
            }
        }

        // ---- elementwise LSTM update + projection (torch gate order i,f,g,o)
        const float* c0p = c0 + myp * 6;
        float c0v[6];
        if (full) {
            v2f a = lo_ldnt2(c0p + 0), b = lo_ldnt2(c0p + 2), c = lo_ldnt2(c0p + 4);
            c0v[0] = a.x; c0v[1] = a.y; c0v[2] = b.x; c0v[3] = b.y; c0v[4] = c.x; c0v[5] = c.y;
        } else {
#pragma unroll
            for (int k = 0; k < 6; ++k) c0v[k] = c0p[k];
        }

        float h1v[6], c1v[6];
        float acc_out = 0.0f;
#pragma unroll
        for (int k = 0; k < 6; ++k) {
            float iv = lo_sigmoid(gate[k]);
            float fv = lo_sigmoid(gate[6 + k]);
            float gv = tanhf(gate[12 + k]);
            float ov = lo_sigmoid(gate[18 + k]);
            float c1 = fv * c0v[k] + iv * gv;
            float h1 = ov * tanhf(c1);
            c1v[k] = c1;
            h1v[k] = h1;
            acc_out += h1 * wo[k];
        }

        float* h1p = out + N + myp * 6;          // h1 block starts after new_p
        float* c1p = out + N * 7 + myp * 6;      // c1 block after h1 (6N)
        if (full) {
            lo_stnt2(h1p + 0, h1v[0], h1v[1]);
            lo_stnt2(h1p + 2, h1v[2], h1v[3]);
            lo_stnt2(h1p + 4, h1v[4], h1v[5]);
            lo_stnt2(c1p + 0, c1v[0], c1v[1]);
            lo_stnt2(c1p + 2, c1v[2], c1v[3]);
            lo_stnt2(c1p + 4, c1v[4], c1v[5]);
        } else {
#pragma unroll
            for (int k = 0; k < 6; ++k) { h1p[k] = h1v[k]; c1p[k] = c1v[k]; }
        }

        float upd = (acc_out + bo) * 0.1f;
        float pv  = lo_ldnt((myp < n1) ? (p1 + myp) : (p2 + (myp - n1)));
        __builtin_nontemporal_store(pv + upd, out + myp);
    }
}

extern "C" void kernel_launch(void* const* d_in, const int* in_sizes, int n_in,
                              void* d_out, int out_size, void* d_ws, size_t ws_size,
                              hipStream_t stream) {
    (void)n_in; (void)out_size; (void)d_ws; (void)ws_size;
    const float* p1    = (const float*)d_in[0];
    const float* p2    = (const float*)d_in[1];
    const float* g1    = (const float*)d_in[2];
    const float* g2    = (const float*)d_in[3];
    const float* h0    = (const float*)d_in[4];
    const float* c0    = (const float*)d_in[5];
    const float* W_ih  = (const float*)d_in[6];
    const float* W_hh  = (const float*)d_in[7];
    const float* b_ih  = (const float*)d_in[8];
    const float* b_hh  = (const float*)d_in[9];
    const float* W_out = (const float*)d_in[10];
    const float* b_out = (const float*)d_in[11];

    long long n1 = in_sizes[0];
    long long n2 = in_sizes[1];
    long long N  = n1 + n2;

    long long groups  = (N + 31) / 32;
    long long waves   = (groups + GROUPS - 1) / GROUPS;
    long long threads = waves * 32;
    int block = 256;
    long long blocks = (threads + block - 1) / block;

    lopt_lstm_step_kernel<<<(dim3)(unsigned)blocks, (dim3)block, 0, stream>>>(
        p1, p2, g1, g2, h0, c0, W_ih, W_hh, b_ih, b_hh, W_out, b_out,
        (float*)d_out, n1, N);
}